// FloodGNN_73899207295430
// MI455X (gfx1250) — compile-verified
//
#include <hip/hip_runtime.h>
#include <hip/hip_bf16.h>
#include <math.h>

#define N_NODES 50000
#define N_EDGES 800000
#define D_IN    37
#define H       64
#define L       3
#define BN_EPS  1e-5f

typedef float v2f __attribute__((ext_vector_type(2)));
typedef float v8f __attribute__((ext_vector_type(8)));

// ---------------------------------------------------------------------------
// Degree / normalization
// ---------------------------------------------------------------------------
__global__ void init_deg_kernel(float* deg, int n) {
    int i = blockIdx.x * blockDim.x + threadIdx.x;
    if (i < n) deg[i] = 1.0f;  // self-loop contributes 1
}

__global__ void accum_deg_kernel(const long long* __restrict__ ei, float* deg, int e) {
    int i = blockIdx.x * blockDim.x + threadIdx.x;
    if (i < e) atomicAdd(&deg[(int)ei[(size_t)N_EDGES + i]], 1.0f);
}

__global__ void dinv_kernel(float* deg, int n) {
    int i = blockIdx.x * blockDim.x + threadIdx.x;
    if (i < n) deg[i] = rsqrtf(deg[i]);  // deg >= 1 always (self loops)
}

// ---------------------------------------------------------------------------
// Input projection: h = relu(x @ Wp + b), K = 37 (odd -> plain VALU kernel)
// ---------------------------------------------------------------------------
__global__ void proj_kernel(const float* __restrict__ x, const float* __restrict__ w,
                            const float* __restrict__ b, float* __restrict__ h, int n) {
    __shared__ float sw[D_IN * H];
    for (int i = threadIdx.x; i < D_IN * H; i += blockDim.x) sw[i] = w[i];
    __syncthreads();
    int idx = blockIdx.x * blockDim.x + threadIdx.x;
    if (idx >= n * H) return;
    int node = idx >> 6;
    int f    = idx & 63;
    const float* xr = x + (size_t)node * D_IN;
    float acc = b[f];
#pragma unroll
    for (int k = 0; k < D_IN; ++k) acc = fmaf(xr[k], sw[k * H + f], acc);
    h[idx] = fmaxf(acc, 0.0f);
}

// ---------------------------------------------------------------------------
// FP32 WMMA GEMM: C[nrows x NT*16] = A[nrows x 64] @ W[64 x NT*16]
// One wave32 owns a 16-row tile; K stepped by 4 via V_WMMA_F32_16X16X4_F32.
// nrows must be a multiple of 16 (50000 = 16*3125 -> always full tiles).
// ---------------------------------------------------------------------------
template <int NT, bool RELU, bool BIAS>
__global__ void gemm_wmma_kernel(const float* __restrict__ A, const float* __restrict__ W,
                                 const float* __restrict__ bias, float* __restrict__ C,
                                 int nrows) {
    const int lane = threadIdx.x & 31;
    const int wave = threadIdx.x >> 5;
    const int tile = blockIdx.x * (blockDim.x >> 5) + wave;
    const int row0 = tile * 16;
    if (row0 >= nrows) return;               // wave-uniform exit: EXEC stays all-1s
    const int half = lane >> 4;              // 0: K even pair, 1: K odd pair
    const int l16  = lane & 15;
    const int OUTC = NT * 16;

    v8f acc[NT];
    const v8f zero = {};
#pragma unroll
    for (int t = 0; t < NT; ++t) acc[t] = zero;

    const float* arow = A + (size_t)(row0 + l16) * 64;
#pragma unroll
    for (int k0 = 0; k0 < 64; k0 += 4) {
        const int kk = k0 + half * 2;
        // A fragment (16x4 f32): lane l16 = row M, half selects K pair
        float2 av = *(const float2*)(arow + kk);
        v2f a; a[0] = av.x; a[1] = av.y;
#pragma unroll
        for (int t = 0; t < NT; ++t) {
            // B fragment (4x16 f32): lane l16 = col N, half selects K pair
            v2f b;
            b[0] = W[(size_t)kk * OUTC + t * 16 + l16];
            b[1] = W[(size_t)(kk + 1) * OUTC + t * 16 + l16];
            acc[t] = __builtin_amdgcn_wmma_f32_16x16x4_f32(
                false, a, false, b, (short)0, acc[t], false, false);
        }
    }
    // Epilogue. C/D layout: VGPR r -> M=r (lanes 0-15), M=r+8 (lanes 16-31); N=lane&15.
#pragma unroll
    for (int t = 0; t < NT; ++t) {
        const int col = t * 16 + l16;
        float bv = BIAS ? bias[col] : 0.0f;
#pragma unroll
        for (int r = 0; r < 8; ++r) {
            int row = row0 + r + half * 8;
            float v = acc[t][r] + bv;
            if (RELU) v = fmaxf(v, 0.0f);
            C[(size_t)row * OUTC + col] = v;
        }
    }
}

// ---------------------------------------------------------------------------
// agg init with conv bias
// ---------------------------------------------------------------------------
__global__ void init_agg_kernel(float* __restrict__ agg, const float* __restrict__ bias,
                                int n64) {
    int idx = blockIdx.x * blockDim.x + threadIdx.x;
    if (idx < n64) agg[idx] = bias[idx & 63];
}

// ---------------------------------------------------------------------------
// Edge gather-scale-scatter: 16 threads per edge, float4 gather + 4 f32 atomics
// Edges [0,E) from edge_index; [E, E+N) are implicit self loops.
// ---------------------------------------------------------------------------
__global__ void edge_scatter_kernel(const long long* __restrict__ ei,
                                    const float* __restrict__ dinv,
                                    const float* __restrict__ hl,
                                    float* __restrict__ agg) {
    long long t = (long long)blockIdx.x * blockDim.x + threadIdx.x;
    const long long total = (long long)(N_EDGES + N_NODES) * 16;
    if (t >= total) return;
    int e = (int)(t >> 4);
    int q = (int)(t & 15);
    int s, d;
    if (e < N_EDGES) {
        s = (int)ei[e];
        d = (int)ei[(size_t)N_EDGES + e];
    } else {
        s = d = e - N_EDGES;
    }
    float nrm = dinv[s] * dinv[d];
    float4 v = *(const float4*)(hl + (size_t)s * 64 + q * 4);
    float* dst = agg + (size_t)d * 64 + q * 4;
    atomicAdd(dst + 0, v.x * nrm);
    atomicAdd(dst + 1, v.y * nrm);
    atomicAdd(dst + 2, v.z * nrm);
    atomicAdd(dst + 3, v.w * nrm);
}

// ---------------------------------------------------------------------------
// BatchNorm: per-feature sum / sumsq reduction (stage 1 + finalize), then apply
// ---------------------------------------------------------------------------
__global__ void zero_stats_kernel(float* stats) {
    if (threadIdx.x < 128) stats[threadIdx.x] = 0.0f;  // [0:64)=sum, [64:128)=sumsq
}

__global__ void bn_stats_kernel(const float* __restrict__ agg, float* __restrict__ stats,
                                int n) {
    __shared__ float ssum[256];
    __shared__ float ssqs[256];
    int f = threadIdx.x & 63;
    int rlane = threadIdx.x >> 6;  // 0..3 row sub-lanes per block sweep
    float s = 0.0f, q = 0.0f;
    for (long long r = (long long)blockIdx.x * 4 + rlane; r < n;
         r += (long long)gridDim.x * 4) {
        float v = agg[r * 64 + f];
        s += v;
        q += v * v;
    }
    ssum[threadIdx.x] = s;
    ssqs[threadIdx.x] = q;
    __syncthreads();
    if (threadIdx.x < 64) {
        s = ssum[threadIdx.x] + ssum[threadIdx.x + 64] + ssum[threadIdx.x + 128] +
            ssum[threadIdx.x + 192];
        q = ssqs[threadIdx.x] + ssqs[threadIdx.x + 64] + ssqs[threadIdx.x + 128] +
            ssqs[threadIdx.x + 192];
        atomicAdd(&stats[f], s);
        atomicAdd(&stats[64 + f], q);
    }
}

__global__ void bn_finalize_kernel(float* __restrict__ stats,
                                   const float* __restrict__ g,
                                   const float* __restrict__ b, int n) {
    int f = threadIdx.x;
    if (f >= 64) return;
    float inv_n = 1.0f / (float)n;
    float mean = stats[f] * inv_n;
    float var  = stats[64 + f] * inv_n - mean * mean;
    float sc   = g[f] * rsqrtf(var + BN_EPS);
    stats[128 + f] = sc;               // scale
    stats[192 + f] = b[f] - mean * sc; // shift
}

__global__ void bn_apply_kernel(const float* __restrict__ agg,
                                const float* __restrict__ stats,
                                float* __restrict__ h, int n64, int residual) {
    int idx = blockIdx.x * blockDim.x + threadIdx.x;
    if (idx >= n64) return;
    int f = idx & 63;
    float v = fmaf(agg[idx], stats[128 + f], stats[192 + f]);
    v = fmaxf(v, 0.0f);
    if (residual) v += h[idx];
    h[idx] = v;  // in-place: pure elementwise, safe
}

// ---------------------------------------------------------------------------
// Final head: out[n] = sigmoid(m2[n] . w3 + b3), m2 is N x 32
// ---------------------------------------------------------------------------
__global__ void head_kernel(const float* __restrict__ m2, const float* __restrict__ w3,
                            const float* __restrict__ b3, float* __restrict__ out, int n) {
    int i = blockIdx.x * blockDim.x + threadIdx.x;
    if (i >= n) return;
    float acc = b3[0];
    const float* r = m2 + (size_t)i * 32;
#pragma unroll
    for (int k = 0; k < 32; ++k) acc = fmaf(r[k], w3[k], acc);
    out[i] = 1.0f / (1.0f + __expf(-acc));
}

// ---------------------------------------------------------------------------
// Host launcher
// ---------------------------------------------------------------------------
extern "C" void kernel_launch(void* const* d_in, const int* in_sizes, int n_in,
                              void* d_out, int out_size, void* d_ws, size_t ws_size,
                              hipStream_t stream) {
    const float*     x      = (const float*)d_in[0];       // (N, 37)
    const long long* ei     = (const long long*)d_in[1];   // (2, E) int64
    const float*     proj_w = (const float*)d_in[2];       // (37, 64)
    const float*     proj_b = (const float*)d_in[3];       // (64,)
    const float*     conv_w = (const float*)d_in[4];       // (3, 64, 64)
    const float*     conv_b = (const float*)d_in[5];       // (3, 64)
    const float*     bn_g   = (const float*)d_in[6];       // (3, 64)
    const float*     bn_b   = (const float*)d_in[7];       // (3, 64)
    const float*     w1     = (const float*)d_in[8];       // (64, 64)
    const float*     b1     = (const float*)d_in[9];       // (64,)
    const float*     w2     = (const float*)d_in[10];      // (64, 32)
    const float*     b2     = (const float*)d_in[11];      // (32,)
    const float*     w3     = (const float*)d_in[12];      // (32, 1)
    const float*     b3     = (const float*)d_in[13];      // (1,)
    float*           out    = (float*)d_out;

    // Workspace layout (floats), offsets padded for 16B alignment
    float* ws    = (float*)d_ws;
    float* deg   = ws;                        // N floats (becomes dinv)
    float* h     = ws + 50048;                // N*64
    float* hl    = h  + (size_t)N_NODES * 64; // N*64 (also m1)
    float* agg   = hl + (size_t)N_NODES * 64; // N*64 (also m2, row stride 32)
    float* stats = agg + (size_t)N_NODES * 64;// 256: sum|sumsq|scale|shift

    const int NH = N_NODES * H;

    // 1) degree -> dinv
    init_deg_kernel<<<(N_NODES + 255) / 256, 256, 0, stream>>>(deg, N_NODES);
    accum_deg_kernel<<<(N_EDGES + 255) / 256, 256, 0, stream>>>(ei, deg, N_EDGES);
    dinv_kernel<<<(N_NODES + 255) / 256, 256, 0, stream>>>(deg, N_NODES);

    // 2) input projection + relu
    proj_kernel<<<(NH + 255) / 256, 256, 0, stream>>>(x, proj_w, proj_b, h, N_NODES);

    // 3) GCN layers
    const int gemm_blocks   = (N_NODES / 16 + 3) / 4;  // 4 waves/block, 16 rows/wave
    const long long e_work  = (long long)(N_EDGES + N_NODES) * 16;
    const int scatter_blocks = (int)((e_work + 255) / 256);
    for (int i = 0; i < L; ++i) {
        gemm_wmma_kernel<4, false, false><<<gemm_blocks, 128, 0, stream>>>(
            h, conv_w + (size_t)i * 64 * 64, nullptr, hl, N_NODES);
        init_agg_kernel<<<(NH + 255) / 256, 256, 0, stream>>>(agg, conv_b + i * 64, NH);
        edge_scatter_kernel<<<scatter_blocks, 256, 0, stream>>>(ei, deg, hl, agg);
        zero_stats_kernel<<<1, 128, 0, stream>>>(stats);
        bn_stats_kernel<<<256, 256, 0, stream>>>(agg, stats, N_NODES);
        bn_finalize_kernel<<<1, 64, 0, stream>>>(stats, bn_g + i * 64, bn_b + i * 64,
                                                 N_NODES);
        bn_apply_kernel<<<(NH + 255) / 256, 256, 0, stream>>>(agg, stats, h, NH,
                                                              i > 0 ? 1 : 0);
    }

    // 4) MLP head
    gemm_wmma_kernel<4, true, true><<<gemm_blocks, 128, 0, stream>>>(h, w1, b1, hl,
                                                                     N_NODES);
    gemm_wmma_kernel<2, true, true><<<gemm_blocks, 128, 0, stream>>>(hl, w2, b2, agg,
                                                                     N_NODES);
    head_kernel<<<(N_NODES + 255) / 256, 256, 0, stream>>>(agg, w3, b3, out, N_NODES);
}